// DistSageConv_68161130987987
// MI455X (gfx1250) — compile-verified
//
#include <hip/hip_runtime.h>

// ---------------------------------------------------------------------------
// GAT (H=1, F=128) for MI455X / gfx1250, wave32.
//   el = X_src . attn_l   (WMMA f32 16x16x4, 16 nodes per wave)
//   er = X_dst . attn_r
//   fused per-destination softmax + weighted aggregation (dst_idx sorted).
// ---------------------------------------------------------------------------

typedef float v2f __attribute__((ext_vector_type(2)));
typedef float v8f __attribute__((ext_vector_type(8)));

#define GAT_F 128
#define NEG_SLOPE 0.2f

// --------------------------------------------------------------------------
// Kernel 1: per-node attention logits via V_WMMA_F32_16X16X4_F32.
// One wave handles a tile of 16 nodes: A = 16x4 slice of node rows,
// B = attn[k..k+3] broadcast across all 16 columns, accumulate K=128 in
// 32 WMMA steps. Column N=0 of D holds the 16 dot products:
//   D vgpr i, lane 0  -> node0 + i      (M = 0..7)
//   D vgpr i, lane 16 -> node0 + 8 + i  (M = 8..15)
// --------------------------------------------------------------------------
__global__ void gat_logits_wmma(const float* __restrict__ x,
                                const float* __restrict__ attn,
                                float* __restrict__ out, int n_nodes) {
  const int lane = threadIdx.x & 31;
  const int wave = threadIdx.x >> 5;
  const int tile = blockIdx.x * (blockDim.x >> 5) + wave;
  const int node0 = tile * 16;
  if (node0 >= n_nodes) return;  // wave-uniform

  const int row = lane & 15;
  const int khalf = (lane >> 4) * 2;  // lanes 0-15 -> K {0,1}; 16-31 -> K {2,3}
  int r = node0 + row;
  if (r >= n_nodes) r = n_nodes - 1;  // clamp (partial tile): duplicate reads ok
  const float* rowp = x + (size_t)r * GAT_F;

  v8f c = {0.f, 0.f, 0.f, 0.f, 0.f, 0.f, 0.f, 0.f};
  #pragma unroll 4
  for (int k0 = 0; k0 < GAT_F; k0 += 4) {
    v2f a, b;
    a.x = rowp[k0 + khalf];
    a.y = rowp[k0 + khalf + 1];
    b.x = attn[k0 + khalf];      // B[K][N] = attn[k0+K] for every column N
    b.y = attn[k0 + khalf + 1];
    c = __builtin_amdgcn_wmma_f32_16x16x4_f32(false, a, false, b,
                                              (short)0, c, false, false);
  }

  if (lane == 0) {
    #pragma unroll
    for (int i = 0; i < 8; ++i)
      if (node0 + i < n_nodes) out[node0 + i] = c[i];
  } else if (lane == 16) {
    #pragma unroll
    for (int i = 0; i < 8; ++i)
      if (node0 + 8 + i < n_nodes) out[node0 + 8 + i] = c[i];
  }
}

// --------------------------------------------------------------------------
// Kernel 2: fused per-destination softmax + weighted aggregation.
// One 128-thread block per destination v. dst_idx is sorted, so the edge
// range [lo,hi) is found by binary search. No atomics; output written once
// (deterministic). Pass A: ex -> LDS + block-reduced denominator.
// Pass B: thread f accumulates sum_e ex_e * x_src[src_e, f].
// --------------------------------------------------------------------------
#define EX_CAP 6144  // 24 KB LDS stash; avg degree ~160, so always hit in practice

__global__ __launch_bounds__(GAT_F) void gat_fused_aggregate(
    const float* __restrict__ x_src, const int* __restrict__ src_idx,
    const int* __restrict__ dst_idx, const float* __restrict__ el,
    const float* __restrict__ er, float* __restrict__ out, long nE) {
  __shared__ float ex_s[EX_CAP];
  __shared__ long range_s[2];
  __shared__ float red_s[GAT_F / 32];
  __shared__ float sum_s;

  const int v = blockIdx.x;
  const int tid = threadIdx.x;  // 0..127 == feature index in pass B

  if (tid == 0) {
    long lo = 0, hi = nE;  // lower_bound(v)
    while (lo < hi) { long m = (lo + hi) >> 1; if (dst_idx[m] < v) lo = m + 1; else hi = m; }
    range_s[0] = lo;
    long lo2 = lo, hi2 = nE;  // upper_bound(v)
    while (lo2 < hi2) { long m = (lo2 + hi2) >> 1; if (dst_idx[m] <= v) lo2 = m + 1; else hi2 = m; }
    range_s[1] = lo2;
  }
  __syncthreads();
  const long lo = range_s[0], hi = range_s[1];
  const long deg = hi - lo;

  if (deg == 0) {  // empty neighborhood: reference segment_sum yields zeros
    out[(size_t)v * GAT_F + tid] = 0.f;
    return;
  }

  const float er_v = er[v];
  const bool fits = (deg <= EX_CAP);

  // ---- Pass A: ex = exp(leaky_relu(el[u] + er[v])), stash + reduce sum ----
  float psum = 0.f;
  for (long e = lo + tid; e < hi; e += GAT_F) {
    float t = el[src_idx[e]] + er_v;
    t = (t >= 0.f) ? t : NEG_SLOPE * t;
    float w = expf(t);
    if (fits) ex_s[e - lo] = w;
    psum += w;
  }
  #pragma unroll
  for (int off = 16; off > 0; off >>= 1) psum += __shfl_down(psum, off, 32);
  const int lane = tid & 31, wid = tid >> 5;
  if (lane == 0) red_s[wid] = psum;
  __syncthreads();
  if (tid == 0) {
    float s = 0.f;
    #pragma unroll
    for (int i = 0; i < GAT_F / 32; ++i) s += red_s[i];
    sum_s = s;
  }
  __syncthreads();
  const float inv = 1.0f / sum_s;

  // ---- Pass B: weighted gather-accumulate, thread = feature lane f ----
  const int f = tid;
  float a0 = 0.f, a1 = 0.f, a2 = 0.f, a3 = 0.f;
  long e = lo;
  if (fits) {
    for (; e + 3 < hi; e += 4) {
      const int s0 = src_idx[e + 0], s1 = src_idx[e + 1];
      const int s2 = src_idx[e + 2], s3 = src_idx[e + 3];
      // prefetch a few edges ahead (global_prefetch_b8)
      long ep = e + 8; if (ep >= hi) ep = hi - 1;
      const int sp = src_idx[ep];
      if ((f & 31) == 0)
        __builtin_prefetch(&x_src[(size_t)sp * GAT_F + f], 0, 3);
      a0 = fmaf(ex_s[e + 0 - lo], x_src[(size_t)s0 * GAT_F + f], a0);
      a1 = fmaf(ex_s[e + 1 - lo], x_src[(size_t)s1 * GAT_F + f], a1);
      a2 = fmaf(ex_s[e + 2 - lo], x_src[(size_t)s2 * GAT_F + f], a2);
      a3 = fmaf(ex_s[e + 3 - lo], x_src[(size_t)s3 * GAT_F + f], a3);
    }
    for (; e < hi; ++e)
      a0 = fmaf(ex_s[e - lo], x_src[(size_t)src_idx[e] * GAT_F + f], a0);
  } else {  // pathological degree: recompute ex instead of stashing
    for (; e < hi; ++e) {
      const int s = src_idx[e];
      float t = el[s] + er_v;
      t = (t >= 0.f) ? t : NEG_SLOPE * t;
      a0 = fmaf(expf(t), x_src[(size_t)s * GAT_F + f], a0);
    }
  }
  const float acc = (a0 + a1) + (a2 + a3);
  out[(size_t)v * GAT_F + f] = acc * inv;
}

// --------------------------------------------------------------------------
// Host launcher
// --------------------------------------------------------------------------
extern "C" void kernel_launch(void* const* d_in, const int* in_sizes, int n_in,
                              void* d_out, int out_size, void* d_ws, size_t ws_size,
                              hipStream_t stream) {
  const float* x_src  = (const float*)d_in[0];
  const float* x_dst  = (const float*)d_in[1];
  const float* attn_l = (const float*)d_in[2];
  const float* attn_r = (const float*)d_in[3];
  const int*   src_idx = (const int*)d_in[4];
  const int*   dst_idx = (const int*)d_in[5];
  float* out = (float*)d_out;

  const int n_src = in_sizes[0] / GAT_F;
  const int n_dst = in_sizes[1] / GAT_F;
  const long nE   = (long)in_sizes[4];

  float* ws = (float*)d_ws;
  float* el = ws;            // n_src floats
  float* er = ws + n_src;    // n_dst floats

  // Phase 1: node logits via WMMA (16 nodes per wave, 8 waves per block)
  const int tiles_src = (n_src + 15) / 16;
  const int tiles_dst = (n_dst + 15) / 16;
  gat_logits_wmma<<<(tiles_src + 7) / 8, 256, 0, stream>>>(x_src, attn_l, el, n_src);
  gat_logits_wmma<<<(tiles_dst + 7) / 8, 256, 0, stream>>>(x_dst, attn_r, er, n_dst);

  // Phase 2: fused per-destination softmax + aggregation (one block per dst)
  gat_fused_aggregate<<<n_dst, GAT_F, 0, stream>>>(x_src, src_idx, dst_idx,
                                                   el, er, out, nE);
}